// LocalSemanticAlignment_56418690400860
// MI455X (gfx1250) — compile-verified
//
#include <hip/hip_runtime.h>
#include <math.h>

typedef float v2f __attribute__((ext_vector_type(2)));
typedef float v8f __attribute__((ext_vector_type(8)));

#define HW    4096      // 64*64 feature locations
#define CF    256       // feature channels
#define C2    64        // unalign_fb channels
#define NH    3         // parse heads (channels 1..3)
#define HWRAW 65536     // 256*256
#define ALPHAV 100.0f

// ---------------- preprocessing ----------------

__global__ __launch_bounds__(256) void mean_kernel(const float* __restrict__ fa,
                                                   const float* __restrict__ fb,
                                                   float* __restrict__ mean_ab) {
  int ch = blockIdx.x;                                // 0..511 (fa then fb)
  const float* src = (ch < CF) ? (fa + ch * HW) : (fb + (ch - CF) * HW);
  __shared__ float red[256];
  float s = 0.f;
  for (int i = threadIdx.x; i < HW; i += 256) s += src[i];
  red[threadIdx.x] = s;
  __syncthreads();
  for (int st = 128; st > 0; st >>= 1) {
    if (threadIdx.x < st) red[threadIdx.x] += red[threadIdx.x + st];
    __syncthreads();
  }
  if (threadIdx.x == 0) mean_ab[ch] = red[0] * (1.0f / HW);
}

__global__ __launch_bounds__(256) void invnorm_kernel(const float* __restrict__ fa,
                                                      const float* __restrict__ fb,
                                                      const float* __restrict__ mean_ab,
                                                      float* __restrict__ rnorm_ab) {
  int idx = blockIdx.x * blockDim.x + threadIdx.x;    // 0..8191
  if (idx >= 2 * HW) return;
  int which = idx >> 12;
  int p = idx & (HW - 1);
  const float* src = which ? fb : fa;
  const float* mn = mean_ab + which * CF;
  float s = 0.f;
  for (int c = 0; c < CF; ++c) { float v = src[c * HW + p] - mn[c]; s += v * v; }
  rnorm_ab[idx] = 1.0f / sqrtf(s);
}

// fan: plain (c, p) layout (staged to LDS per p-tile).
// fbn: channel-PAIR-interleaved layout fbnp[(c>>1)*(2*HW) + p*2 + (c&1)] so the
//      WMMA A-fragment (two adjacent channels, same q) is ONE aligned b64 load.
__global__ __launch_bounds__(256) void normalize_kernel(const float* __restrict__ fa,
                                                        const float* __restrict__ fb,
                                                        const float* __restrict__ mean_ab,
                                                        const float* __restrict__ rnorm_ab,
                                                        float* __restrict__ fan,
                                                        float* __restrict__ fbnp) {
  int idx = blockIdx.x * blockDim.x + threadIdx.x;
  if (idx >= 2 * CF * HW) return;
  int which = idx / (CF * HW);
  int r = idx - which * (CF * HW);
  int c = r / HW, p = r & (HW - 1);
  const float* src = which ? fb : fa;
  float v = (src[r] - mean_ab[which * CF + c]) * rnorm_ab[which * HW + p];
  if (which)
    fbnp[(size_t)(c >> 1) * (2 * HW) + p * 2 + (c & 1)] = v;
  else
    fan[r] = v;
}

__global__ __launch_bounds__(256) void mask_kernel(const int* __restrict__ fa_parse,
                                                   const int* __restrict__ fb_parse,
                                                   float* __restrict__ mask_a,
                                                   float* __restrict__ mask_b) {
  int idx = blockIdx.x * blockDim.x + threadIdx.x;    // 0..2*3*4096-1
  if (idx >= 2 * NH * HW) return;
  int which = idx / (NH * HW);
  int r = idx - which * (NH * HW);
  int k = r / HW, p = r & (HW - 1);
  int y = p >> 6, x = p & 63;
  // nearest 256->64: src = dst*4
  const int* parse = which ? fb_parse : fa_parse;
  int v = parse[(k + 1) * HWRAW + (y * 4) * 256 + (x * 4)];
  (which ? mask_b : mask_a)[r] = (v != 0) ? 1.0f : 0.0f;
}

__global__ __launch_bounds__(256) void rnormq_kernel(const float* __restrict__ mask_a,
                                                     float* __restrict__ rnormq) {
  int p = blockIdx.x * blockDim.x + threadIdx.x;
  if (p >= HW) return;
  float s = mask_a[p] + mask_a[HW + p] + mask_a[2 * HW + p];
  rnormq[p] = 1.0f / fmaxf(s, 1.0f);
}

__global__ __launch_bounds__(256) void ufb_kernel(const float* __restrict__ ufb_raw,
                                                  float* __restrict__ ufb) {
  int idx = blockIdx.x * blockDim.x + threadIdx.x;    // 64*4096
  if (idx >= C2 * HW) return;
  int c = idx / HW, p = idx & (HW - 1);
  int y = p >> 6, x = p & 63;
  const float sc = 255.0f / 63.0f;                    // align_corners 256->64
  float ry = (float)y * sc, rx = (float)x * sc;
  int y0 = (int)floorf(ry), x0 = (int)floorf(rx);
  float wy = ry - (float)y0, wx = rx - (float)x0;
  int y1 = min(y0 + 1, 255), x1 = min(x0 + 1, 255);
  const float* base = ufb_raw + c * HWRAW;
  float v00 = base[y0 * 256 + x0], v01 = base[y0 * 256 + x1];
  float v10 = base[y1 * 256 + x0], v11 = base[y1 * 256 + x1];
  float r0 = v00 * (1.f - wy) + v10 * wy;
  float r1 = v01 * (1.f - wy) + v11 * wy;
  ufb[idx] = r0 * (1.f - wx) + r1 * wx;
}

// ---------------- fused flash-attention core (f32 WMMA) ----------------
// One wave handles (head k, 16-query p-tile, q-chunk). Online softmax over q.
// GEMM1: E(16q x 16p) = fbn_tile^T x fan_tile via V_WMMA_F32_16X16X4_F32 (64 K-steps)
// GEMM2: acc(16c2 x 16p) += V^T x P, 4 c2-chunks x 4 K-steps per q-tile.

__global__ __launch_bounds__(32) void attn_kernel(const float* __restrict__ fan,
                                                  const float* __restrict__ fbnp,
                                                  const float* __restrict__ ufb,
                                                  const float* __restrict__ mask_b,
                                                  float* __restrict__ pacc,
                                                  float* __restrict__ pml,
                                                  int qsplit) {
  __shared__ float sfa[16 * 260];   // fan p-tile, transposed + padded (bank-conflict free b64)
  __shared__ float smb[HW];         // mask_b for this wave's q-range

  int b = blockIdx.x;
  int qc = b % qsplit;
  int tmp = b / qsplit;
  int pt = tmp & 255;
  int khead = tmp >> 8;
  int p0 = pt * 16;
  int qlen = HW / qsplit;
  int qbeg = qc * qlen;

  int lane = threadIdx.x;
  int half = lane >> 4;             // 0: K=0,1 / M rows 0..7 ; 1: K=2,3 / M rows 8..15
  int lp = lane & 15;

  // stage fan tile (256 x 16) into LDS: sfa[col*260 + c]
  for (int i = lane; i < 16 * CF; i += 32) {
    int col = i & 15, c = i >> 4;
    sfa[col * 260 + c] = fan[c * HW + p0 + col];
  }
  const float* mbp = mask_b + khead * HW + qbeg;
  for (int i = lane; i < qlen; i += 32) smb[i] = mbp[i];
  __syncthreads();

  float m = -3.0e38f;
  float l = 0.f;
  v8f acc[4] = {};

  // running lane pointers (advance, never recompute in the loop)
  const float* fA = fbnp + (size_t)(qbeg + lp) * 2 + (size_t)half * (2 * HW);
  const float* uP = ufb + (size_t)lp * HW + qbeg + 2 * half;

  for (int qq = 0; qq < qlen; qq += 16) {
    // ---- GEMM1: E = fbn_q^T * fan_p, contraction over 256 channels, K=4/step
    v8f e = {};
#pragma unroll 8
    for (int c0 = 0; c0 < CF; c0 += 4) {
      // paired layout: pair index = c0/2 (+half folded into fA)
      v2f a = *reinterpret_cast<const v2f*>(&fA[(size_t)(c0 >> 1) * (2 * HW)]);
      int ca = c0 + 2 * half;
      v2f bv = *reinterpret_cast<const v2f*>(&sfa[lp * 260 + ca]);
      e = __builtin_amdgcn_wmma_f32_16x16x4_f32(false, a, false, bv, (short)0, e,
                                                false, false);
    }
    // ---- scale by alpha * mask_b (mask_a==1 rows unaffected; mask_a==0 cols dropped at merge)
    float tmax = -3.0e38f;
#pragma unroll
    for (int j = 0; j < 8; ++j) {
      float mb = smb[qq + j + 8 * half];          // row q = q0 + j + 8*half
      float v = e[j] * (ALPHAV * mb);
      e[j] = v;
      tmax = fmaxf(tmax, v);
    }
    tmax = fmaxf(tmax, __shfl_xor(tmax, 16));
    float mnew = fmaxf(m, tmax);
    float corr = __expf(m - mnew);
    m = mnew;
    v8f P;
    float ls = 0.f;
#pragma unroll
    for (int j = 0; j < 8; ++j) {
      float pe = __expf(e[j] - mnew);
      P[j] = pe;
      ls += pe;
    }
    ls += __shfl_xor(ls, 16);
    l = l * corr + ls;
#pragma unroll
    for (int ch = 0; ch < 4; ++ch)
#pragma unroll
      for (int j = 0; j < 8; ++j) acc[ch][j] *= corr;

    // shadow copy of P from the other lane-half (q rows 8..15 <-> 0..7)
    v8f Po;
#pragma unroll
    for (int j = 0; j < 8; ++j) Po[j] = __shfl_xor(P[j], 16);

    // ---- GEMM2: acc(16c2 x 16p) += ufb^T(16c2 x 4q) * P(4q x 16p), 4 K-steps
#pragma unroll
    for (int t = 0; t < 4; ++t) {
      v2f bb;                      // B rows q = 4t + 2*half + {0,1}, col = lp
      if (t < 2) {
        bb.x = half ? Po[4 * t + 2] : P[4 * t];
        bb.y = half ? Po[4 * t + 3] : P[4 * t + 1];
      } else {
        bb.x = half ? P[4 * t - 6] : Po[4 * t - 8];
        bb.y = half ? P[4 * t - 5] : Po[4 * t - 7];
      }
#pragma unroll
      for (int ch = 0; ch < 4; ++ch) {
        // immediate offsets: ch*16*HW + 4t floats (<= 768KB+48B, fits 24-bit ioffset)
        v2f av = *reinterpret_cast<const v2f*>(&uP[(size_t)ch * (16 * HW) + 4 * t]);
        acc[ch] = __builtin_amdgcn_wmma_f32_16x16x4_f32(false, av, false, bb, (short)0,
                                                        acc[ch], false, false);
      }
    }
    fA += 32;   // q advances by 16, paired layout stride 2
    uP += 16;
  }

  // ---- write partial (raw acc + per-column m,l) for log-sum-exp merge
  float* pa = pacc + (size_t)b * (C2 * 16);
#pragma unroll
  for (int ch = 0; ch < 4; ++ch)
#pragma unroll
    for (int j = 0; j < 8; ++j) {
      int c2r = ch * 16 + j + 8 * half;           // D layout: M = reg + 8*half
      pa[c2r * 16 + lp] = acc[ch][j];
    }
  if (half == 0) {
    pml[b * 32 + lp] = m;
    pml[b * 32 + 16 + lp] = l;
  }
}

__global__ __launch_bounds__(256) void merge_kernel(const float* __restrict__ pacc,
                                                    const float* __restrict__ pml,
                                                    const float* __restrict__ mask_a,
                                                    float* __restrict__ headout,
                                                    int qsplit) {
  int b = blockIdx.x;                // (khead*256 + pt)
  int pt = b & 255, khead = b >> 8;
  int t = threadIdx.x;
  int col = t & 15;
  int c2 = t >> 4;                   // 0..15; handles c2 + {0,16,32,48}
  int base = b * qsplit;

  float M = -3.0e38f;
  for (int i = 0; i < qsplit; ++i) M = fmaxf(M, pml[(base + i) * 32 + col]);
  float w[4];
  float L = 0.f;
  for (int i = 0; i < qsplit; ++i) {
    float wi = __expf(pml[(base + i) * 32 + col] - M);
    w[i] = wi;
    L += pml[(base + i) * 32 + 16 + col] * wi;
  }
  int p = pt * 16 + col;
  float scale = mask_a[khead * HW + p] / L;
  for (int cc = 0; cc < 4; ++cc) {
    int c2r = c2 + cc * 16;
    float s = 0.f;
    for (int i = 0; i < qsplit; ++i)
      s += pacc[(size_t)(base + i) * (C2 * 16) + c2r * 16 + col] * w[i];
    headout[(size_t)khead * C2 * HW + c2r * HW + p] = s * scale;
  }
}

// ---------------- final bilinear upsample 64x64 -> 256x256 ----------------

__global__ __launch_bounds__(256) void upsample_kernel(const float* __restrict__ headout,
                                                       const float* __restrict__ rnormq,
                                                       float* __restrict__ out) {
  int idx = blockIdx.x * blockDim.x + threadIdx.x;
  if (idx >= C2 * HWRAW) return;
  int c = idx >> 16;
  int r = idx & (HWRAW - 1);
  int Y = r >> 8, X = r & 255;
  const float sc = 63.0f / 255.0f;   // align_corners 64->256
  float ry = (float)Y * sc, rx = (float)X * sc;
  int y0 = (int)ry, x0 = (int)rx;
  float wy = ry - (float)y0, wx = rx - (float)x0;
  int y1 = min(y0 + 1, 63), x1 = min(x0 + 1, 63);
  const float* h0 = headout + c * HW;
  const float* h1 = h0 + C2 * HW;
  const float* h2 = h1 + C2 * HW;
  int p00 = y0 * 64 + x0, p01 = y0 * 64 + x1, p10 = y1 * 64 + x0, p11 = y1 * 64 + x1;
  float a00 = (h0[p00] + h1[p00] + h2[p00]) * rnormq[p00];
  float a01 = (h0[p01] + h1[p01] + h2[p01]) * rnormq[p01];
  float a10 = (h0[p10] + h1[p10] + h2[p10]) * rnormq[p10];
  float a11 = (h0[p11] + h1[p11] + h2[p11]) * rnormq[p11];
  float r0 = a00 * (1.f - wy) + a10 * wy;
  float r1 = a01 * (1.f - wy) + a11 * wy;
  out[idx] = r0 * (1.f - wx) + r1 * wx;
}

// ---------------- host launcher ----------------

extern "C" void kernel_launch(void* const* d_in, const int* in_sizes, int n_in,
                              void* d_out, int out_size, void* d_ws, size_t ws_size,
                              hipStream_t stream) {
  const float* unalign_fb = (const float*)d_in[0];  // (1,64,256,256)
  const float* fa = (const float*)d_in[1];          // (1,256,64,64)
  const int* fa_parse = (const int*)d_in[2];        // (1,4,256,256)
  const float* fb = (const float*)d_in[3];          // (1,256,64,64)
  const int* fb_parse = (const int*)d_in[4];        // (1,4,256,256)
  float* out = (float*)d_out;                       // (1,64,256,256)

  // workspace layout (floats); pacc last so layout is qsplit-independent
  float* ws = (float*)d_ws;
  size_t off = 0;
  float* fan = ws + off;      off += (size_t)CF * HW;     // 1,048,576
  float* fbnp = ws + off;     off += (size_t)CF * HW;     // 1,048,576 (pair-interleaved)
  float* ufb = ws + off;      off += (size_t)C2 * HW;     //   262,144
  float* mask_a = ws + off;   off += (size_t)NH * HW;     //    12,288
  float* mask_b = ws + off;   off += (size_t)NH * HW;     //    12,288
  float* mean_ab = ws + off;  off += 2 * CF;              //       512
  float* rnorm_ab = ws + off; off += 2 * HW;              //     8,192
  float* rnormq = ws + off;   off += HW;                  //     4,096
  float* headout = ws + off;  off += (size_t)NH * C2 * HW;//   786,432
  float* pml = ws + off;      off += (size_t)NH * 256 * 4 * 32;
  float* pacc = ws + off;     // NH*256*qsplit * (64*16)

  size_t need4 = (off + (size_t)NH * 256 * 4 * C2 * 16) * sizeof(float);
  int qsplit = (ws_size >= need4) ? 4 : 1;

  mean_kernel<<<2 * CF, 256, 0, stream>>>(fa, fb, mean_ab);
  invnorm_kernel<<<(2 * HW + 255) / 256, 256, 0, stream>>>(fa, fb, mean_ab, rnorm_ab);
  normalize_kernel<<<(2 * CF * HW + 255) / 256, 256, 0, stream>>>(fa, fb, mean_ab,
                                                                  rnorm_ab, fan, fbnp);
  mask_kernel<<<(2 * NH * HW + 255) / 256, 256, 0, stream>>>(fa_parse, fb_parse,
                                                             mask_a, mask_b);
  rnormq_kernel<<<(HW + 255) / 256, 256, 0, stream>>>(mask_a, rnormq);
  ufb_kernel<<<(C2 * HW + 255) / 256, 256, 0, stream>>>(unalign_fb, ufb);

  attn_kernel<<<NH * 256 * qsplit, 32, 0, stream>>>(fan, fbnp, ufb, mask_b, pacc, pml,
                                                    qsplit);
  merge_kernel<<<NH * 256, 256, 0, stream>>>(pacc, pml, mask_a, headout, qsplit);
  upsample_kernel<<<(C2 * HWRAW + 255) / 256, 256, 0, stream>>>(headout, rnormq, out);
}